// GraphAutoencoderNeighborhood_77867757076501
// MI455X (gfx1250) — compile-verified
//
#include <hip/hip_runtime.h>
#include <math.h>

// ---------------------------------------------------------------------------
// Graph VAE (GAT encoder/decoder) forward for gfx1250.
// Dense projections use V_WMMA_F32_16X16X4_F32 (full f32 on the matrix pipe).
// Edge softmax/aggregation use streaming atomics (global_atomic_add_f32 and
// order-preserving-uint atomicMax for the segment max).
// ---------------------------------------------------------------------------

typedef float v2f __attribute__((ext_vector_type(2)));
typedef float v8f __attribute__((ext_vector_type(8)));

#define WPB 8           // waves (16x16 output tiles along M) per block
#define SLOPE 0.2f
#define BN_EPS 1e-5f

// ----- order-preserving float<->uint encoding for atomic max ---------------
__device__ __forceinline__ unsigned fenc(float f) {
    unsigned u = __float_as_uint(f);
    return (u & 0x80000000u) ? ~u : (u | 0x80000000u);
}
__device__ __forceinline__ float fdec(unsigned u) {
    return (u & 0x80000000u) ? __uint_as_float(u & 0x7fffffffu)
                             : __uint_as_float(~u);
}
// enc(-inf) = ~0xFF800000 = 0x007FFFFF
#define ENC_NEG_INF 0x007FFFFFu

// ===========================================================================
// Generic f32 WMMA GEMM: C[M,N] = A[M,K] @ B[K,N] (+bias) (+relu)
// One wave computes a 16x16 tile; K stepped by 4 with v_wmma_f32_16x16x4_f32.
// A-frag (16x4): lane l -> row l&15, VGPR0/1 = K = 2*(l>>4) + {0,1}.
// B-frag (4x16): lane l -> col l&15, VGPR0/1 = K = 2*(l>>4) + {0,1}.
// C/D (16x16):  VGPR r at lane l -> row r + 8*(l>>4), col l&15.
// For columns n >= N the load address is clamped to column N-1: the WMMA then
// computes finite garbage in those output columns, which the store epilogue
// never writes. This keeps the K-loop branch-free (no EXEC churn) and lets
// the addresses strength-reduce to pure pointer increments.
// ===========================================================================
__global__ void gemm_wmma_f32(const float* __restrict__ A, const float* __restrict__ B,
                              const float* __restrict__ bias, float* __restrict__ C,
                              int M, int K, int N, int act)
{
    const int wave = threadIdx.x >> 5;
    const int lane = threadIdx.x & 31;
    const int half = lane >> 4;
    const int lm   = lane & 15;
    const int m0   = (blockIdx.x * WPB + wave) << 4;
    const int n0   = blockIdx.y << 4;
    if (m0 >= M) return;                    // wave-uniform: EXEC stays all-1s

    v8f acc = {0.f, 0.f, 0.f, 0.f, 0.f, 0.f, 0.f, 0.f};
    const int n  = n0 + lm;
    const int nc = n < N ? n : (N - 1);     // safe (in-bounds) load column

    const float* ap = A + (size_t)(m0 + lm) * K + (half << 1);
    const float* bp = B + (size_t)(half << 1) * N + nc;
    const size_t bstep = (size_t)4 * N;

    for (int k0 = 0; k0 < K; k0 += 4) {
        v2f a, b;
        a.x = ap[0];
        a.y = ap[1];
        b.x = bp[0];
        b.y = bp[N];
        acc = __builtin_amdgcn_wmma_f32_16x16x4_f32(
                  false, a, false, b, (short)0, acc, false, false);
        ap += 4;
        bp += bstep;
    }

    if (n < N) {
        const float bv = bias ? bias[n] : 0.f;
#pragma unroll
        for (int r = 0; r < 8; ++r) {
            const int m = m0 + r + (half << 3);
            float v = acc[r] + bv;
            if (act == 1) v = fmaxf(v, 0.f);
            C[(size_t)m * N + n] = v;
        }
    }
}

// ===========================================================================
// Utility fills
// ===========================================================================
__global__ void fill_f32(float* p, float v, size_t n) {
    size_t i = (size_t)blockIdx.x * blockDim.x + threadIdx.x;
    if (i < n) p[i] = v;
}
__global__ void fill_u32(unsigned* p, unsigned v, size_t n) {
    size_t i = (size_t)blockIdx.x * blockDim.x + threadIdx.x;
    if (i < n) p[i] = v;
}

// ===========================================================================
// Node-side attention coefficients: s_src[n,h] = <h[n,h,:], a_s[h,:]>, same a_d
// ===========================================================================
__global__ void att_scores(const float* __restrict__ h, const float* __restrict__ a_s,
                           const float* __restrict__ a_d, float* __restrict__ ssrc,
                           float* __restrict__ sdst, int N, int H, int C)
{
    int n = blockIdx.x * blockDim.x + threadIdx.x;
    if (n >= N) return;
    const float* hp = h + (size_t)n * H * C;
    for (int hh = 0; hh < H; ++hh) {
        float as = 0.f, ad = 0.f;
        for (int c = 0; c < C; ++c) {
            const float v = hp[hh * C + c];
            as += v * a_s[hh * C + c];
            ad += v * a_d[hh * C + c];
        }
        ssrc[(size_t)n * H + hh] = as;
        sdst[(size_t)n * H + hh] = ad;
    }
}

// ===========================================================================
// Edge pass 1: e = leaky_relu(ssrc[src]+sdst[dst]); segment max via atomicMax
// Edges e<E come from edge_index; e>=E are the appended self-loops.
// ===========================================================================
__global__ void edge_max(const int* __restrict__ ei, int E, int N,
                         const float* __restrict__ ssrc, const float* __restrict__ sdst,
                         unsigned* __restrict__ menc, int H)
{
    int e = blockIdx.x * blockDim.x + threadIdx.x;
    const int Etot = E + N;
    if (e >= Etot) return;
    int s, d;
    if (e < E) { s = ei[e]; d = ei[E + e]; } else { s = e - E; d = s; }
    for (int h = 0; h < H; ++h) {
        float v = ssrc[(size_t)s * H + h] + sdst[(size_t)d * H + h];
        v = v > 0.f ? v : SLOPE * v;
        atomicMax(&menc[(size_t)d * H + h], fenc(v));
    }
}

// ===========================================================================
// Edge pass 2: ex = exp(e - max[dst]); store ex; segment sum via atomicAdd
// ===========================================================================
__global__ void edge_expsum(const int* __restrict__ ei, int E, int N,
                            const float* __restrict__ ssrc, const float* __restrict__ sdst,
                            const unsigned* __restrict__ menc, float* __restrict__ denom,
                            float* __restrict__ exbuf, int H)
{
    int e = blockIdx.x * blockDim.x + threadIdx.x;
    const int Etot = E + N;
    if (e >= Etot) return;
    int s, d;
    if (e < E) { s = ei[e]; d = ei[E + e]; } else { s = e - E; d = s; }
    for (int h = 0; h < H; ++h) {
        float v = ssrc[(size_t)s * H + h] + sdst[(size_t)d * H + h];
        v = v > 0.f ? v : SLOPE * v;
        const float ex = __expf(v - fdec(menc[(size_t)d * H + h]));
        exbuf[(size_t)e * H + h] = ex;
        atomicAdd(&denom[(size_t)d * H + h], ex);
    }
}

// ===========================================================================
// Edge pass 3: out[dst] += (ex/denom[dst]) * h[src]  -- one wave per edge,
// lanes stride the H*C feature row (coalesced gather + coalesced atomics).
// C is a power of two; Cshift = log2(C).
// ===========================================================================
__global__ void edge_aggregate(const int* __restrict__ ei, int E, int N,
                               const float* __restrict__ hbuf,
                               const float* __restrict__ exbuf,
                               const float* __restrict__ denom,
                               float* __restrict__ out, int H, int Cshift)
{
    const int HC = H << Cshift;
    const size_t gtid = (size_t)blockIdx.x * blockDim.x + threadIdx.x;
    const int e = (int)(gtid >> 5);
    const int lane = threadIdx.x & 31;
    const int Etot = E + N;
    if (e >= Etot) return;
    int s, d;
    if (e < E) { s = ei[e]; d = ei[E + e]; } else { s = e - E; d = s; }
    for (int f = lane; f < HC; f += 32) {
        const int h = f >> Cshift;
        const float w = exbuf[(size_t)e * H + h] / denom[(size_t)d * H + h];
        atomicAdd(&out[(size_t)d * HC + f], w * hbuf[(size_t)s * HC + f]);
    }
}

// ===========================================================================
// BatchNorm: stats (sum, sumsq per channel) then apply (+relu, +optional res)
// D is a power of two and divides 256; gridDim.x*256 is a multiple of D so
// each thread owns a fixed channel and reads coalesced.
// ===========================================================================
__global__ void bn_stats(const float* __restrict__ x, float* __restrict__ sums,
                         int N, int D)
{
    const int t = blockIdx.x * blockDim.x + threadIdx.x;
    const int nth = gridDim.x * blockDim.x;
    const int d = t & (D - 1);
    float s = 0.f, s2 = 0.f;
    const size_t total = (size_t)N * D;
    for (size_t i = t; i < total; i += nth) { const float v = x[i]; s += v; s2 += v * v; }
    atomicAdd(&sums[d], s);
    atomicAdd(&sums[D + d], s2);
}

__global__ void bn_apply(const float* __restrict__ x, const float* __restrict__ sums,
                         const float* __restrict__ g, const float* __restrict__ b,
                         const float* __restrict__ residual, float* __restrict__ y,
                         int N, int D)
{
    const size_t i = (size_t)blockIdx.x * blockDim.x + threadIdx.x;
    if (i >= (size_t)N * D) return;
    const int d = (int)(i & (size_t)(D - 1));
    const float mean = sums[d] / (float)N;
    const float var  = sums[D + d] / (float)N - mean * mean;
    float v = (x[i] - mean) * rsqrtf(var + BN_EPS) * g[d] + b[d];
    v = fmaxf(v, 0.f);
    if (residual) v += residual[i];
    y[i] = v;
}

// ===========================================================================
// Misc fused kernels
// ===========================================================================
__global__ void embed_concat(const float* __restrict__ xf, const int* __restrict__ xid,
                             const float* __restrict__ emb, float* __restrict__ x0, int N)
{
    int n = blockIdx.x * blockDim.x + threadIdx.x;
    if (n >= N) return;
    const float* e = emb + (size_t)xid[n] * 8;
    const float* f = xf + (size_t)n * 8;
    float* o = x0 + (size_t)n * 16;
#pragma unroll
    for (int c = 0; c < 8; ++c) o[c] = e[c];
#pragma unroll
    for (int c = 0; c < 8; ++c) o[8 + c] = f[c];
}

__global__ void z_kl(const float* __restrict__ mu, const float* __restrict__ lv,
                     const float* __restrict__ eps, float* __restrict__ zws,
                     float* __restrict__ zout, float* __restrict__ klsum, size_t n)
{
    __shared__ float red[256];
    const size_t i = (size_t)blockIdx.x * 256 + threadIdx.x;
    float c = 0.f;
    if (i < n) {
        const float m = mu[i], l = lv[i];
        const float zv = m + eps[i] * __expf(0.5f * l);
        zws[i] = zv;
        zout[i] = zv;
        c = 1.f + l - m * m - __expf(l);
    }
    red[threadIdx.x] = c;
    __syncthreads();
    for (int s = 128; s > 0; s >>= 1) {
        if (threadIdx.x < s) red[threadIdx.x] += red[threadIdx.x + s];
        __syncthreads();
    }
    if (threadIdx.x == 0) atomicAdd(klsum, red[0]);
}

__global__ void kl_finalize(const float* __restrict__ klsum, float* __restrict__ out, float inv)
{
    if (blockIdx.x == 0 && threadIdx.x == 0) out[0] = -0.5f * klsum[0] * inv;
}

__global__ void sigmoid_bias(const float* __restrict__ x, const float* __restrict__ b,
                             float* __restrict__ y, int N, int D)
{
    const size_t i = (size_t)blockIdx.x * blockDim.x + threadIdx.x;
    if (i >= (size_t)N * D) return;
    const float v = x[i] + b[i & (size_t)(D - 1)];
    y[i] = 1.f / (1.f + __expf(-v));
}

// ===========================================================================
// Host-side orchestration
// ===========================================================================
static void launch_gemm(hipStream_t st, const float* A, const float* B, const float* bias,
                        float* C, int M, int K, int N, int act)
{
    dim3 g((M + 16 * WPB - 1) / (16 * WPB), (N + 15) / 16);
    gemm_wmma_f32<<<g, 256, 0, st>>>(A, B, bias, C, M, K, N, act);
}

struct GatBufs {
    float* hbuf; float* ssrc; float* sdst; unsigned* menc;
    float* denom; float* exbuf; float* agg;
};

static void run_gat(hipStream_t st, const float* X, int K, const float* W,
                    const float* a_s, const float* a_d, int H, int C, int Cshift,
                    const int* ei, int E, int N, const GatBufs& bf)
{
    const int HC = H * C;
    const int Etot = E + N;
    launch_gemm(st, X, W, nullptr, bf.hbuf, N, K, HC, 0);
    att_scores<<<(N + 255) / 256, 256, 0, st>>>(bf.hbuf, a_s, a_d, bf.ssrc, bf.sdst, N, H, C);
    fill_u32<<<((size_t)N * H + 255) / 256, 256, 0, st>>>(bf.menc, ENC_NEG_INF, (size_t)N * H);
    fill_f32<<<((size_t)N * H + 255) / 256, 256, 0, st>>>(bf.denom, 0.f, (size_t)N * H);
    fill_f32<<<((size_t)N * HC + 255) / 256, 256, 0, st>>>(bf.agg, 0.f, (size_t)N * HC);
    edge_max<<<(Etot + 255) / 256, 256, 0, st>>>(ei, E, N, bf.ssrc, bf.sdst, bf.menc, H);
    edge_expsum<<<(Etot + 255) / 256, 256, 0, st>>>(ei, E, N, bf.ssrc, bf.sdst, bf.menc,
                                                    bf.denom, bf.exbuf, H);
    const size_t aggth = (size_t)Etot * 32;
    edge_aggregate<<<(aggth + 255) / 256, 256, 0, st>>>(ei, E, N, bf.hbuf, bf.exbuf,
                                                        bf.denom, bf.agg, H, Cshift);
}

static void run_bn(hipStream_t st, const float* x, const float* g, const float* b,
                   const float* residual, float* y, float* sums, int N, int D)
{
    fill_f32<<<1, 256, 0, st>>>(sums, 0.f, (size_t)(2 * D));
    bn_stats<<<64, 256, 0, st>>>(x, sums, N, D);
    bn_apply<<<((size_t)N * D + 255) / 256, 256, 0, st>>>(x, sums, g, b, residual, y, N, D);
}

extern "C" void kernel_launch(void* const* d_in, const int* in_sizes, int n_in,
                              void* d_out, int out_size, void* d_ws, size_t ws_size,
                              hipStream_t stream)
{
    // ---- inputs (setup_inputs dict order, params dict flattened in order) ----
    const float* x_feat = (const float*)d_in[0];       // [N,8]
    const int*   x_id   = (const int*)d_in[1];         // [N]
    const int*   ei     = (const int*)d_in[2];         // [2,E] row-major
    const float* eps    = (const float*)d_in[4];       // [N,32]
    const float* id_emb = (const float*)d_in[5];       // [256,8]
    const float* e1_W  = (const float*)d_in[6],  *e1_as = (const float*)d_in[7],
               * e1_ad = (const float*)d_in[8];
    const float* e2_W  = (const float*)d_in[10], *e2_as = (const float*)d_in[11],
               * e2_ad = (const float*)d_in[12];
    const float* e3_W  = (const float*)d_in[14], *e3_as = (const float*)d_in[15],
               * e3_ad = (const float*)d_in[16];
    const float* d1_W  = (const float*)d_in[18], *d1_as = (const float*)d_in[19],
               * d1_ad = (const float*)d_in[20];
    const float* d2_W  = (const float*)d_in[22], *d2_as = (const float*)d_in[23],
               * d2_ad = (const float*)d_in[24];
    const float* d3_W  = (const float*)d_in[26], *d3_as = (const float*)d_in[27],
               * d3_ad = (const float*)d_in[28], *d3_b = (const float*)d_in[29];
    const float* bn1_g = (const float*)d_in[30], *bn1_b = (const float*)d_in[31];
    const float* bn2_g = (const float*)d_in[32], *bn2_b = (const float*)d_in[33];
    const float* bn3_g = (const float*)d_in[34], *bn3_b = (const float*)d_in[35];
    const float* dbn1_g = (const float*)d_in[36], *dbn1_b = (const float*)d_in[37];
    const float* dbn2_g = (const float*)d_in[38], *dbn2_b = (const float*)d_in[39];
    const float* zm_W = (const float*)d_in[40], *zm_b = (const float*)d_in[41];
    const float* zl_W = (const float*)d_in[42], *zl_b = (const float*)d_in[43];
    const float* cls_W = (const float*)d_in[44], *cls_b = (const float*)d_in[45];
    const float* nb_W1 = (const float*)d_in[46], *nb_b1 = (const float*)d_in[47];
    const float* nb_W2 = (const float*)d_in[48], *nb_b2 = (const float*)d_in[49];
    const float* nb_W3 = (const float*)d_in[50], *nb_b3 = (const float*)d_in[51];

    const int N = in_sizes[0] / 8;
    const int E = in_sizes[2] / 2;
    const int Etot = E + N;

    float* out = (float*)d_out;
    // output layout: cont[N,8] | canid[N,256] | neighbor[N,256] | z[N,32] | kl
    float* out_cont  = out;
    float* out_canid = out + (size_t)N * 8;
    float* out_nbr   = out_canid + (size_t)N * 256;
    float* out_z     = out_nbr + (size_t)N * 256;
    float* out_kl    = out_z + (size_t)N * 32;

    // ---- workspace bump allocator (floats) ----
    float* wsf = (float*)d_ws;
    size_t off = 0;
    auto alloc = [&](size_t n) { float* p = wsf + off; off += (n + 63) & ~(size_t)63; return p; };

    float* x0     = alloc((size_t)N * 16);
    float* x1     = alloc((size_t)N * 128);   // reused later as d1 output
    float* x2     = alloc((size_t)N * 32);
    float* hsum   = alloc((size_t)N * 32);
    float* mu     = alloc((size_t)N * 32);    // reused later as nb t1
    float* logv   = alloc((size_t)N * 32);    // reused later as nb t2
    float* zws    = alloc((size_t)N * 32);
    float* d2o    = alloc((size_t)N * 32);
    float* hbuf   = alloc((size_t)N * 128);
    float* agg    = alloc((size_t)N * 128);
    float* ssrc   = alloc((size_t)N * 4);
    float* sdst   = alloc((size_t)N * 4);
    unsigned* menc = (unsigned*)alloc((size_t)N * 4);
    float* denom  = alloc((size_t)N * 4);
    float* exbuf  = alloc((size_t)Etot * 4);
    float* sums   = alloc(256);
    float* klsum  = alloc(64);

    GatBufs bf = { hbuf, ssrc, sdst, menc, denom, exbuf, agg };

    // ---- encoder ----
    embed_concat<<<(N + 255) / 256, 256, 0, stream>>>(x_feat, x_id, id_emb, x0, N);

    run_gat(stream, x0, 16, e1_W, e1_as, e1_ad, 4, 32, 5, ei, E, N, bf);   // -> agg [N,128]
    run_bn(stream, agg, bn1_g, bn1_b, nullptr, x1, sums, N, 128);

    run_gat(stream, x1, 128, e2_W, e2_as, e2_ad, 1, 32, 5, ei, E, N, bf);  // -> agg [N,32]
    run_bn(stream, agg, bn2_g, bn2_b, nullptr, x2, sums, N, 32);

    run_gat(stream, x2, 32, e3_W, e3_as, e3_ad, 1, 32, 5, ei, E, N, bf);   // -> agg [N,32]
    run_bn(stream, agg, bn3_g, bn3_b, x2, hsum, sums, N, 32);              // h = x3 + x2

    launch_gemm(stream, hsum, zm_W, zm_b, mu,   N, 32, 32, 0);
    launch_gemm(stream, hsum, zl_W, zl_b, logv, N, 32, 32, 0);

    fill_f32<<<1, 64, 0, stream>>>(klsum, 0.f, 64);
    z_kl<<<((size_t)N * 32 + 255) / 256, 256, 0, stream>>>(mu, logv, eps, zws, out_z,
                                                           klsum, (size_t)N * 32);

    // ---- node decoder ----
    run_gat(stream, zws, 32, d1_W, d1_as, d1_ad, 4, 32, 5, ei, E, N, bf);  // -> agg [N,128]
    run_bn(stream, agg, dbn1_g, dbn1_b, nullptr, x1, sums, N, 128);        // x1 := d1 out

    run_gat(stream, x1, 128, d2_W, d2_as, d2_ad, 1, 32, 5, ei, E, N, bf);  // -> agg [N,32]
    run_bn(stream, agg, dbn2_g, dbn2_b, nullptr, d2o, sums, N, 32);

    run_gat(stream, d2o, 32, d3_W, d3_as, d3_ad, 1, 8, 3, ei, E, N, bf);   // -> agg [N,8]
    sigmoid_bias<<<((size_t)N * 8 + 255) / 256, 256, 0, stream>>>(agg, d3_b, out_cont, N, 8);

    launch_gemm(stream, d2o, cls_W, cls_b, out_canid, N, 32, 256, 0);

    // ---- neighborhood decoder MLP (t1/t2 alias mu/logv, which are dead) ----
    launch_gemm(stream, zws, nb_W1, nb_b1, mu,   N, 32, 32, 1);
    launch_gemm(stream, mu,  nb_W2, nb_b2, logv, N, 32, 32, 1);
    launch_gemm(stream, logv, nb_W3, nb_b3, out_nbr, N, 32, 256, 0);

    kl_finalize<<<1, 1, 0, stream>>>(klsum, out_kl, 1.f / ((float)N * 32.f));

    (void)n_in; (void)out_size; (void)ws_size;
}